// test_network_47167330845370
// MI455X (gfx1250) — compile-verified
//
#include <hip/hip_runtime.h>
#include <math.h>
#include <stdint.h>

// ---------------------------------------------------------------------------
// Per-row map, memory-bound: 268 MB @ 23.3 TB/s => ~11.5 us floor.
// Strategy: 1 row = 1 float4; NT b128 load -> ~40 VALU ops -> NT b128 store.
// Params (36 floats) staged into LDS via CDNA5 async global->LDS DMA.
// ---------------------------------------------------------------------------

typedef float v4f __attribute__((ext_vector_type(4)));

#define TPB 256
#define ROWS_PER_THREAD 4

// p += q @ S  (upper shear), row = [p0 p1 q0 q1]
__device__ __forceinline__ void lin_up(float r[4], const float s[3]) {
    r[0] = fmaf(r[3], s[1], fmaf(r[2], s[0], r[0]));
    r[1] = fmaf(r[3], s[2], fmaf(r[2], s[1], r[1]));
}
// q += p @ S  (lower shear)
__device__ __forceinline__ void lin_low(float r[4], const float s[3]) {
    r[2] = fmaf(r[1], s[1], fmaf(r[0], s[0], r[2]));
    r[3] = fmaf(r[1], s[2], fmaf(r[0], s[1], r[3]));
}
// double application of PERM [1,3,0,2] == full reverse [3,2,1,0]
__device__ __forceinline__ void rev(float r[4]) {
    float t0 = r[0], t1 = r[1];
    r[0] = r[3]; r[1] = r[2]; r[2] = t1; r[3] = t0;
}

__global__ __launch_bounds__(TPB) void symplectic_chain_kernel(
    const float* __restrict__ x,
    const float* __restrict__ luA,   // 16 floats: 4 x (2x2)
    const float* __restrict__ aup,   // 2 floats
    const float* __restrict__ llA,   // 16 floats: 4 x (2x2)
    const float* __restrict__ alow,  // 2 floats
    float* __restrict__ out,
    int nrows)
{
    // LDS staging buffer: bytes [0,64)=luA, [64,128)=llA, [128,136)=aup,
    // [136,144)=alow.
    __shared__ __align__(16) float sp[48];

    const int tid = threadIdx.x;

    // --- CDNA5 async global->LDS parameter broadcast (ASYNCcnt path) -------
    // Low 32 bits of a generic LDS pointer are the LDS byte address.
    unsigned lds_lo = (unsigned)(uintptr_t)(void*)sp;
    if (tid < 4) {
        unsigned voff = (unsigned)tid * 16u;
        unsigned dA = lds_lo + voff;          // luA -> sp[0..15]
        asm volatile("global_load_async_to_lds_b128 %0, %1, %2 offset:0"
                     :: "v"(dA), "v"(voff), "s"(luA) : "memory");
        unsigned dB = lds_lo + 64u + voff;    // llA -> sp[16..31]
        asm volatile("global_load_async_to_lds_b128 %0, %1, %2 offset:0"
                     :: "v"(dB), "v"(voff), "s"(llA) : "memory");
    }
    if (tid == 0) {
        unsigned z = 0u;
        asm volatile("global_load_async_to_lds_b64 %0, %1, %2 offset:0"
                     :: "v"(lds_lo + 128u), "v"(z), "s"(aup) : "memory");
        asm volatile("global_load_async_to_lds_b64 %0, %1, %2 offset:0"
                     :: "v"(lds_lo + 136u), "v"(z), "s"(alow) : "memory");
    }
#if __has_builtin(__builtin_amdgcn_s_wait_asynccnt)
    __builtin_amdgcn_s_wait_asynccnt(0);
#else
    asm volatile("s_wait_asynccnt 0" ::: "memory");
#endif
    __syncthreads();

    // --- hoist params into registers; fold S = A + A^T -> (2A00, A01+A10, 2A11)
    float s[8][3];
#pragma unroll
    for (int m = 0; m < 8; ++m) {
        const float* A = &sp[m * 4];     // m<4: luA[m], m>=4: llA[m-4]
        s[m][0] = 2.0f * A[0];
        s[m][1] = A[1] + A[2];
        s[m][2] = 2.0f * A[3];
    }
    const float a_up0  = sp[32], a_up1  = sp[33];
    const float a_low0 = sp[34], a_low1 = sp[35];

    const v4f* __restrict__ xv = (const v4f*)x;
    v4f* __restrict__ ov = (v4f*)out;

    const int stride = gridDim.x * TPB;
    for (int i = blockIdx.x * TPB + tid; i < nrows; i += stride) {
        v4f in = __builtin_nontemporal_load(&xv[i]);   // b128 NT load
        float r[4] = { in.x, in.y, in.z, in.w };

        rev(r);                                        // _p twice
        lin_up (r, s[0]);                              // lu_A chain
        lin_low(r, s[1]);
        lin_up (r, s[2]);
        lin_low(r, s[3]);
        r[0] += 1.0f; r[1] += 1.0f; r[2] += 1.0f; r[3] += 1.0f;
        rev(r);
        r[0] = fmaf(a_up0, tanhf(r[2]), r[0]);         // _act_up
        r[1] = fmaf(a_up1, tanhf(r[3]), r[1]);
        rev(r);
        lin_low(r, s[4]);                              // ll_A chain
        lin_up (r, s[5]);
        lin_low(r, s[6]);
        lin_up (r, s[7]);
        r[0] += 1.0f; r[1] += 1.0f; r[2] += 1.0f; r[3] += 1.0f;
        rev(r);
        r[2] = fmaf(a_low0, tanhf(r[0]), r[2]);        // _act_low
        r[3] = fmaf(a_low1, tanhf(r[1]), r[3]);
        rev(r);

        v4f o = { r[0], r[1], r[2], r[3] };
        __builtin_nontemporal_store(o, &ov[i]);        // b128 NT store
    }
}

extern "C" void kernel_launch(void* const* d_in, const int* in_sizes, int n_in,
                              void* d_out, int out_size, void* d_ws, size_t ws_size,
                              hipStream_t stream) {
    (void)n_in; (void)out_size; (void)d_ws; (void)ws_size;
    const float* x    = (const float*)d_in[0];
    const float* luA  = (const float*)d_in[1];
    const float* aup  = (const float*)d_in[2];
    const float* llA  = (const float*)d_in[3];
    const float* alow = (const float*)d_in[4];
    float* out = (float*)d_out;

    const int nrows = in_sizes[0] / 4;                       // 8,388,608
    const int threads_needed = (nrows + ROWS_PER_THREAD - 1) / ROWS_PER_THREAD;
    const int blocks = (threads_needed + TPB - 1) / TPB;     // 8192 blocks

    symplectic_chain_kernel<<<blocks, TPB, 0, stream>>>(
        x, luA, aup, llA, alow, out, nrows);
}